// MNIST_CNN2D_TWN_42597485641955
// MI455X (gfx1250) — compile-verified
//
#include <hip/hip_runtime.h>
#include <hip/hip_bf16.h>

typedef __attribute__((ext_vector_type(2))) float v2f;
typedef __attribute__((ext_vector_type(8))) float v8f;

#define TWN 0.05f
#define EPS 1e-5f
#define NB  4096

// ---- workspace layout (float offsets) ----
#define OFF_ACC1   0u         // 64  (sum[32], sumsq[32])
#define OFF_ACC2   64u        // 128 (sum[64], sumsq[64])
#define OFF_TMAX   192u       // 4 uint slots
#define OFF_SS1    256u       // 64  (scale[32], shift[32])
#define OFF_SS2    320u       // 128
#define OFF_SS3    448u       // 1024 (scale[512], shift[512])
#define OFF_Q1     1536u      // 800
#define OFF_B2     2560u      // 51200   B2[k][o], k=c*25+ky*5+kx
#define OFF_BF1    53760u     // 524288  Bf1[k'][j], k'=(r*64+c)
#define OFF_BF2    578048u    // 8192    Bf2[k][n] padded n<16
#define OFF_H1     2097152u   // 18874368  [4096][32][12][12] NCHW
#define OFF_Y2     20971520u  // 16777216  [4096][8][8][64] NHWC
#define OFF_H2     37748736u  // 4194304   [4096][1024]
#define OFF_Z1     41943040u  // 2097152   [4096][512]
#define OFF_A2     44040192u  // 2097152   [4096][512]

__device__ __forceinline__ float tern(float v, float t) {
    return (v > t ? 1.0f : 0.0f) - (v < -t ? 1.0f : 0.0f);
}

#define WMMA_F32X4(a, b, c) \
    __builtin_amdgcn_wmma_f32_16x16x4_f32(false, (a), false, (b), (short)0, (c), false, false)

// ---------- setup kernels ----------
__global__ void zero_small(float* ws) {
    if (threadIdx.x < 196) ws[threadIdx.x] = 0.0f;  // acc1, acc2, tmax
}

__global__ void maxabs_kernel(const float* __restrict__ w, int n, unsigned* tmax) {
    __shared__ float red[256];
    int tid = threadIdx.x;
    float m = 0.0f;
    for (int i = blockIdx.x * 256 + tid; i < n; i += gridDim.x * 256)
        m = fmaxf(m, fabsf(w[i]));
    red[tid] = m; __syncthreads();
    for (int off = 128; off > 0; off >>= 1) {
        if (tid < off) red[tid] = fmaxf(red[tid], red[tid + off]);
        __syncthreads();
    }
    if (tid == 0) atomicMax(tmax, __float_as_uint(red[0]));  // |w| >= 0 so uint cmp == float cmp
}

__global__ void quant_w1(const float* __restrict__ w, const unsigned* tmax, float* q) {
    int i = blockIdx.x * 256 + threadIdx.x;
    if (i < 800) q[i] = tern(w[i], TWN * __uint_as_float(tmax[0]));
}
__global__ void quant_w2T(const float* __restrict__ w, const unsigned* tmax, float* B) {
    int i = blockIdx.x * 256 + threadIdx.x;   // i = o*800 + k
    if (i < 51200) {
        int o = i / 800, k = i - o * 800;
        B[k * 64 + o] = tern(w[i], TWN * __uint_as_float(tmax[1]));
    }
}
__global__ void quant_wf1T(const float* __restrict__ w, const unsigned* tmax, float* B) {
    int i = blockIdx.x * 256 + threadIdx.x;   // i = j*1024 + e, e = c*16 + r (NCHW flatten)
    if (i < 524288) {
        int j = i >> 10, e = i & 1023;
        int c = e >> 4, r = e & 15;           // our A uses k' = r*64 + c (NHWC flatten)
        B[(r * 64 + c) * 512 + j] = tern(w[i], TWN * __uint_as_float(tmax[2]));
    }
}
__global__ void quant_wf2T(const float* __restrict__ w, const unsigned* tmax, float* B) {
    int i = blockIdx.x * 256 + threadIdx.x;   // B[k][n] padded to 16 cols
    if (i < 8192) {
        int k = i >> 4, n = i & 15;
        B[i] = (n < 10) ? tern(w[n * 512 + k], TWN * __uint_as_float(tmax[3])) : 0.0f;
    }
}

// ---------- conv1: stats pass (no store) ----------
__global__ void conv1_stats(const float* __restrict__ x, const float* __restrict__ q1,
                            const float* __restrict__ b1, float* acc) {
    __shared__ float w[25];
    __shared__ float red[512];
    int tid = threadIdx.x, c = blockIdx.y;
    if (tid < 25) w[tid] = q1[c * 25 + tid];
    __syncthreads();
    int idx = blockIdx.x * 256 + tid;         // < 4096*576
    int n = idx / 576, rem = idx - n * 576;
    int oy = rem / 24, ox = rem - oy * 24;
    const float* xp = x + (size_t)n * 784 + oy * 28 + ox;
    float s = b1[c];
#pragma unroll
    for (int ky = 0; ky < 5; ky++)
#pragma unroll
        for (int kx = 0; kx < 5; kx++)
            s = fmaf(w[ky * 5 + kx], xp[ky * 28 + kx], s);
    red[tid] = s; red[256 + tid] = s * s; __syncthreads();
    for (int off = 128; off > 0; off >>= 1) {
        if (tid < off) { red[tid] += red[tid + off]; red[256 + tid] += red[256 + tid + off]; }
        __syncthreads();
    }
    if (tid == 0) { atomicAdd(&acc[c], red[0]); atomicAdd(&acc[32 + c], red[256]); }
}

__global__ void bn_finalize(const float* __restrict__ acc, const float* __restrict__ g,
                            const float* __restrict__ be, float* ss, int C, float inv_count) {
    int c = threadIdx.x;
    if (c < C) {
        float mean = acc[c] * inv_count;
        float var  = acc[C + c] * inv_count - mean * mean;
        float sc   = g[c] * rsqrtf(var + EPS);
        ss[c] = sc; ss[C + c] = be[c] - mean * sc;
    }
}

// ---------- conv1 recompute + BN + ReLU + pool -> h1 [N,32,12,12] ----------
__global__ void conv1_bn_pool(const float* __restrict__ x, const float* __restrict__ q1,
                              const float* __restrict__ b1, const float* __restrict__ ss1,
                              float* __restrict__ h1) {
    __shared__ float w[25];
    int tid = threadIdx.x, c = blockIdx.y;
    if (tid < 25) w[tid] = q1[c * 25 + tid];
    __syncthreads();
    int idx = blockIdx.x * 256 + tid;         // < 4096*144
    int n = idx / 144, rem = idx - n * 144;
    int py = rem / 12, px = rem - py * 12;
    float sc = ss1[c], sh = ss1[32 + c], bias = b1[c];
    float mx = -3.4e38f;
#pragma unroll
    for (int dy = 0; dy < 2; dy++)
#pragma unroll
        for (int dx = 0; dx < 2; dx++) {
            int oy = 2 * py + dy, ox = 2 * px + dx;
            const float* xp = x + (size_t)n * 784 + oy * 28 + ox;
            float s = bias;
#pragma unroll
            for (int ky = 0; ky < 5; ky++)
#pragma unroll
                for (int kx = 0; kx < 5; kx++)
                    s = fmaf(w[ky * 5 + kx], xp[ky * 28 + kx], s);
            mx = fmaxf(mx, fmaxf(sc * s + sh, 0.0f));
        }
    h1[((size_t)(n * 32 + c) * 12 + py) * 12 + px] = mx;
}

// ---------- conv2 as implicit-im2col WMMA GEMM, 16x64 tile per wave ----------
// y2 NHWC [N,8,8,64]; one wave covers all 64 output channels of 16 patches.
__global__ void conv2_wmma(const float* __restrict__ h1, const float* __restrict__ B2,
                           const float* __restrict__ b2, float* __restrict__ y2) {
    int wave = (blockIdx.x * 256 + threadIdx.x) >> 5;   // 16384 M-tiles
    int lane = threadIdx.x & 31;
    int tm = wave;
    int m = tm * 16 + (lane & 15);
    int img = m >> 6, p = m & 63;
    int oy = p >> 3, ox = p & 7;
    const float* abase = h1 + (size_t)img * (32 * 144) + oy * 12 + ox;
    int col = lane & 15;
    int khi = (lane >> 4) << 1;       // 0 or 2
    v8f acc0 = {}, acc1 = {}, acc2 = {}, acc3 = {};
#pragma unroll 2
    for (int k0 = 0; k0 < 800; k0 += 4) {
        int kk = k0 + khi;
        int c0 = kk / 25, r0 = kk - c0 * 25, ky0 = r0 / 5, kx0 = r0 - ky0 * 5;
        int k1 = kk + 1;
        int c1 = k1 / 25, r1 = k1 - c1 * 25, ky1 = r1 / 5, kx1 = r1 - ky1 * 5;
        v2f a;
        a.x = abase[c0 * 144 + ky0 * 12 + kx0];
        a.y = abase[c1 * 144 + ky1 * 12 + kx1];
        const float* brow  = B2 + kk * 64 + col;
        const float* brow1 = B2 + k1 * 64 + col;
        v2f b0, b1, b2v, b3;
        b0.x  = brow[0];  b0.y  = brow1[0];
        b1.x  = brow[16]; b1.y  = brow1[16];
        b2v.x = brow[32]; b2v.y = brow1[32];
        b3.x  = brow[48]; b3.y  = brow1[48];
        acc0 = WMMA_F32X4(a, b0,  acc0);
        acc1 = WMMA_F32X4(a, b1,  acc1);
        acc2 = WMMA_F32X4(a, b2v, acc2);
        acc3 = WMMA_F32X4(a, b3,  acc3);
    }
    float bv0 = b2[col], bv1 = b2[col + 16], bv2 = b2[col + 32], bv3 = b2[col + 48];
    int mbase = tm * 16 + ((lane >> 4) << 3);
#pragma unroll
    for (int g = 0; g < 8; ++g) {
        float* out = y2 + (size_t)(mbase + g) * 64 + col;
        out[0]  = acc0[g] + bv0;
        out[16] = acc1[g] + bv1;
        out[32] = acc2[g] + bv2;
        out[48] = acc3[g] + bv3;
    }
}

// ---------- per-channel stats over NHWC [*, 64] ----------
__global__ void stats64(const float* __restrict__ y, float* acc) {
    __shared__ float red[512];
    int tid = threadIdx.x;
    size_t idx = (size_t)blockIdx.x * 256 + tid;
    size_t stride = (size_t)gridDim.x * 256;   // multiple of 64 -> channel constant per thread
    float s = 0.0f, s2 = 0.0f;
    for (; idx < 16777216u; idx += stride) { float v = y[idx]; s += v; s2 += v * v; }
    red[tid] = s; red[256 + tid] = s2; __syncthreads();
    if (tid < 64) {
        float a = red[tid] + red[tid + 64] + red[tid + 128] + red[tid + 192];
        float b = red[256 + tid] + red[320 + tid] + red[384 + tid] + red[448 + tid];
        atomicAdd(&acc[tid], a); atomicAdd(&acc[64 + tid], b);
    }
}

// ---------- BN2 + ReLU + pool: y2 NHWC -> h2 [N,1024] (k' = (py*4+px)*64 + c) ----------
__global__ void bn2_relu_pool(const float* __restrict__ y2, const float* __restrict__ ss2,
                              float* __restrict__ h2) {
    int idx = blockIdx.x * 256 + threadIdx.x;  // < 4096*1024
    int n = idx >> 10, kk = idx & 1023;
    int r = kk >> 6, c = kk & 63;
    int py = r >> 2, px = r & 3;
    float sc = ss2[c], sh = ss2[64 + c];
    const float* base = y2 + ((size_t)(n * 8 + 2 * py) * 8 + 2 * px) * 64 + c;
    float v0 = fmaxf(sc * base[0]   + sh, 0.0f);
    float v1 = fmaxf(sc * base[64]  + sh, 0.0f);
    float v2 = fmaxf(sc * base[512] + sh, 0.0f);
    float v3 = fmaxf(sc * base[576] + sh, 0.0f);
    h2[idx] = fmaxf(fmaxf(v0, v1), fmaxf(v2, v3));
}

// ---------- WMMA GEMM, 16x64 tile per wave: C = A[M][K] * B[K][N] + bias ----------
__global__ void gemm_wmma4(const float* __restrict__ A, int lda,
                           const float* __restrict__ B, int ldb,
                           const float* __restrict__ bias,
                           float* __restrict__ C, int ldc, int K, int ngrp) {
    int wave = (blockIdx.x * 256 + threadIdx.x) >> 5;
    int lane = threadIdx.x & 31;
    int tm = wave / ngrp, tg = wave - tm * ngrp;     // tg covers 64 columns
    int m = tm * 16 + (lane & 15);
    int col = tg * 64 + (lane & 15);
    int khi = (lane >> 4) << 1;
    const float2* ar2 = reinterpret_cast<const float2*>(A + (size_t)m * lda + khi);
    v8f acc0 = {}, acc1 = {}, acc2 = {}, acc3 = {};
#pragma unroll 2
    for (int k0 = 0; k0 < K; k0 += 4) {
        float2 av = ar2[k0 >> 1];
        v2f a; a.x = av.x; a.y = av.y;
        const float* brow  = B + (size_t)(k0 + khi) * ldb + col;
        const float* brow1 = brow + ldb;
        v2f b0, b1, b2v, b3;
        b0.x  = brow[0];  b0.y  = brow1[0];
        b1.x  = brow[16]; b1.y  = brow1[16];
        b2v.x = brow[32]; b2v.y = brow1[32];
        b3.x  = brow[48]; b3.y  = brow1[48];
        acc0 = WMMA_F32X4(a, b0,  acc0);
        acc1 = WMMA_F32X4(a, b1,  acc1);
        acc2 = WMMA_F32X4(a, b2v, acc2);
        acc3 = WMMA_F32X4(a, b3,  acc3);
    }
    float bv0 = bias[col], bv1 = bias[col + 16], bv2 = bias[col + 32], bv3 = bias[col + 48];
    int mbase = tm * 16 + ((lane >> 4) << 3);
#pragma unroll
    for (int g = 0; g < 8; ++g) {
        float* out = C + (size_t)(mbase + g) * ldc + col;
        out[0]  = acc0[g] + bv0;
        out[16] = acc1[g] + bv1;
        out[32] = acc2[g] + bv2;
        out[48] = acc3[g] + bv3;
    }
}

// ---------- generic single-tile WMMA GEMM (used for fc2, N padded to 16) ----------
__global__ void gemm_wmma(const float* __restrict__ A, int lda,
                          const float* __restrict__ B, int ldb,
                          const float* __restrict__ bias,
                          float* __restrict__ C, int ldc,
                          int K, int ntiles, int nvalid) {
    int wave = (blockIdx.x * 256 + threadIdx.x) >> 5;
    int lane = threadIdx.x & 31;
    int tm = wave / ntiles, tn = wave - tm * ntiles;
    int m = tm * 16 + (lane & 15);
    int ncol = tn * 16 + (lane & 15);
    int khi = (lane >> 4) << 1;
    const float2* ar2 = reinterpret_cast<const float2*>(A + (size_t)m * lda + khi);
    v8f acc = {};
#pragma unroll 2
    for (int k0 = 0; k0 < K; k0 += 4) {
        int kk = k0 + khi;
        float2 av = ar2[k0 >> 1];
        v2f a, b;
        a.x = av.x; a.y = av.y;
        b.x = B[(size_t)kk * ldb + ncol];
        b.y = B[(size_t)(kk + 1) * ldb + ncol];
        acc = WMMA_F32X4(a, b, acc);
    }
    int mbase = tm * 16 + ((lane >> 4) << 3);
    if (ncol < nvalid) {
        float bv = bias[ncol];
#pragma unroll
        for (int g = 0; g < 8; ++g)
            C[(size_t)(mbase + g) * ldc + ncol] = acc[g] + bv;
    }
}

// ---------- fc1 BN1d stats + finalize (one block per column) ----------
__global__ void fc1_stats(const float* __restrict__ z1, const float* __restrict__ g3,
                          const float* __restrict__ be3, float* ss3) {
    __shared__ float red[512];
    int j = blockIdx.x, tid = threadIdx.x;
    float s = 0.0f, s2 = 0.0f;
    for (int r = tid; r < 4096; r += 256) {
        float v = z1[(size_t)r * 512 + j]; s += v; s2 += v * v;
    }
    red[tid] = s; red[256 + tid] = s2; __syncthreads();
    for (int off = 128; off > 0; off >>= 1) {
        if (tid < off) { red[tid] += red[tid + off]; red[256 + tid] += red[256 + tid + off]; }
        __syncthreads();
    }
    if (tid == 0) {
        float mean = red[0] * (1.0f / 4096.0f);
        float var  = red[256] * (1.0f / 4096.0f) - mean * mean;
        float sc   = g3[j] * rsqrtf(var + EPS);
        ss3[j] = sc; ss3[512 + j] = be3[j] - mean * sc;
    }
}

__global__ void bn3_relu(const float* __restrict__ z1, const float* __restrict__ ss3,
                         float* __restrict__ a2) {
    int idx = blockIdx.x * 256 + threadIdx.x;  // < 4096*512
    int j = idx & 511;
    a2[idx] = fmaxf(ss3[j] * z1[idx] + ss3[512 + j], 0.0f);
}

extern "C" void kernel_launch(void* const* d_in, const int* in_sizes, int n_in,
                              void* d_out, int out_size, void* d_ws, size_t ws_size,
                              hipStream_t stream) {
    const float* x   = (const float*)d_in[0];
    const float* w1  = (const float*)d_in[1];
    const float* b1  = (const float*)d_in[2];
    const float* g1  = (const float*)d_in[3];
    const float* be1 = (const float*)d_in[4];
    const float* w2  = (const float*)d_in[5];
    const float* b2  = (const float*)d_in[6];
    const float* g2  = (const float*)d_in[7];
    const float* be2 = (const float*)d_in[8];
    const float* wf1 = (const float*)d_in[9];
    const float* bf1 = (const float*)d_in[10];
    const float* g3  = (const float*)d_in[11];
    const float* be3 = (const float*)d_in[12];
    const float* wf2 = (const float*)d_in[13];
    const float* bf2 = (const float*)d_in[14];

    float* ws = (float*)d_ws;
    float*    acc1 = ws + OFF_ACC1;
    float*    acc2 = ws + OFF_ACC2;
    unsigned* tmax = (unsigned*)(ws + OFF_TMAX);
    float*    ss1  = ws + OFF_SS1;
    float*    ss2  = ws + OFF_SS2;
    float*    ss3  = ws + OFF_SS3;
    float*    q1   = ws + OFF_Q1;
    float*    B2   = ws + OFF_B2;
    float*    Bf1  = ws + OFF_BF1;
    float*    Bf2  = ws + OFF_BF2;
    float*    h1   = ws + OFF_H1;
    float*    y2   = ws + OFF_Y2;
    float*    h2   = ws + OFF_H2;
    float*    z1   = ws + OFF_Z1;
    float*    a2   = ws + OFF_A2;

    zero_small<<<1, 256, 0, stream>>>(ws);

    maxabs_kernel<<<4,   256, 0, stream>>>(w1,  800,    tmax + 0);
    maxabs_kernel<<<64,  256, 0, stream>>>(w2,  51200,  tmax + 1);
    maxabs_kernel<<<256, 256, 0, stream>>>(wf1, 524288, tmax + 2);
    maxabs_kernel<<<8,   256, 0, stream>>>(wf2, 5120,   tmax + 3);

    quant_w1  <<<4,    256, 0, stream>>>(w1,  tmax, q1);
    quant_w2T <<<200,  256, 0, stream>>>(w2,  tmax, B2);
    quant_wf1T<<<2048, 256, 0, stream>>>(wf1, tmax, Bf1);
    quant_wf2T<<<32,   256, 0, stream>>>(wf2, tmax, Bf2);

    conv1_stats<<<dim3(9216, 32), 256, 0, stream>>>(x, q1, b1, acc1);
    bn_finalize<<<1, 64, 0, stream>>>(acc1, g1, be1, ss1, 32, 1.0f / 2359296.0f);
    conv1_bn_pool<<<dim3(2304, 32), 256, 0, stream>>>(x, q1, b1, ss1, h1);

    conv2_wmma<<<2048, 256, 0, stream>>>(h1, B2, b2, y2);           // 16384 waves, 16x64/wave
    stats64<<<1024, 256, 0, stream>>>(y2, acc2);
    bn_finalize<<<1, 64, 0, stream>>>(acc2, g2, be2, ss2, 64, 1.0f / 262144.0f);
    bn2_relu_pool<<<16384, 256, 0, stream>>>(y2, ss2, h2);

    gemm_wmma4<<<256, 256, 0, stream>>>(h2, 1024, Bf1, 512, bf1, z1, 512, 1024, 8);
    fc1_stats<<<512, 256, 0, stream>>>(z1, g3, be3, ss3);
    bn3_relu<<<8192, 256, 0, stream>>>(z1, ss3, a2);

    gemm_wmma<<<32, 256, 0, stream>>>(a2, 512, Bf2, 16, bf2, (float*)d_out, 10, 512, 1, 10);
}